// SimLoss_17875653886257
// MI455X (gfx1250) — compile-verified
//
#include <hip/hip_runtime.h>
#include <stdint.h>

// Problem geometry (fixed by reference: B=16, C=3, H=W=512)
#define R_IMG   48
#define H_IMG   512
#define W_IMG   512
#define L_IMG   (H_IMG * W_IMG)        /* 262144 elems per image   */
#define WPR     (W_IMG / 32)           /* 16 mask words per row    */
#define WPI     (H_IMG * WPR)          /* 8192 mask words per img  */
#define EBLK    256                    /* elems per count/compact block */
#define NBLK    (L_IMG / EBLK)         /* 1024 blocks per image    */
#define TOTBLK  (R_IMG * NBLK)         /* 49152 blocks total       */
#define CH      1024                   /* elems per eval chunk     */
#define NEVAL   (R_IMG * (L_IMG / CH)) /* 12288 eval blocks        */

__device__ __forceinline__ float softplusf(float z) {
    // logaddexp(0, z) = max(z,0) + log1p(exp(-|z|))
    return fmaxf(z, 0.0f) + log1pf(expf(-fabsf(z)));
}

__device__ __forceinline__ unsigned ballot32(bool p) {
    return __builtin_amdgcn_ballot_w32(p);
}

// ---------------------------------------------------------------------------
// K1: pack (t>0) and (x>0) into bitmasks via wave32 ballot. One wave == one word.
// ---------------------------------------------------------------------------
__global__ void kpack(const float* __restrict__ x, const float* __restrict__ t,
                      uint32_t* __restrict__ posbits, uint32_t* __restrict__ xbits) {
    int g    = blockIdx.x * 256 + threadIdx.x;
    int lane = threadIdx.x & 31;
    float tv = t[g];
    float xv = x[g];
    unsigned bp = ballot32(tv > 0.0f);
    unsigned bx = ballot32(xv > 0.0f);
    if (lane == 0) {
        posbits[g >> 5] = bp;
        xbits[g >> 5]   = bx;
    }
}

// ---------------------------------------------------------------------------
// K2: 7x7 binary dilation of posbits (== blur(t*255)>0 with reflect padding,
// which collapses to clipping the window to in-range rows/cols).
// ---------------------------------------------------------------------------
__global__ void kdilate(const uint32_t* __restrict__ posbits,
                        uint32_t* __restrict__ augbits) {
    int idx = blockIdx.x * 256 + threadIdx.x;     // flat word id, grid exact
    int r    = idx >> 13;                          // / WPI
    int wIdx = idx & (WPI - 1);
    int h    = wIdx >> 4;                          // / WPR
    int wi   = wIdx & (WPR - 1);
    int base = r * WPI;
    uint32_t acc = 0u;
    #pragma unroll
    for (int dh = -3; dh <= 3; ++dh) {
        int hh = h + dh;
        if (hh < 0 || hh >= H_IMG) continue;
        int rowb   = base + hh * WPR;
        uint32_t c = posbits[rowb + wi];
        uint32_t l = (wi > 0)       ? posbits[rowb + wi - 1] : 0u;
        uint32_t rr = (wi < WPR - 1) ? posbits[rowb + wi + 1] : 0u;
        uint32_t hd = c;
        hd |= (c << 1) | (l >> 31);
        hd |= (c << 2) | (l >> 30);
        hd |= (c << 3) | (l >> 29);
        hd |= (c >> 1) | (rr << 31);
        hd |= (c >> 2) | (rr << 30);
        hd |= (c >> 3) | (rr << 29);
        acc |= hd;
    }
    augbits[idx] = acc;
}

// ---------------------------------------------------------------------------
// K3: per-256-element-block popcounts of pos / falsepos / neg masks.
// One wave per block; lanes 0..7 each popcount one mask word.
// ---------------------------------------------------------------------------
__global__ void kcount(const uint32_t* __restrict__ posbits,
                       const uint32_t* __restrict__ xbits,
                       const uint32_t* __restrict__ augbits,
                       uint32_t* __restrict__ posCnt,
                       uint32_t* __restrict__ fpCnt,
                       uint32_t* __restrict__ negCnt) {
    int b    = blockIdx.x;          // element-block id: words b*8 .. b*8+7
    int lane = threadIdx.x;         // 0..31
    uint32_t p = 0, f = 0, n = 0;
    if (lane < 8) {
        int w = b * 8 + lane;
        uint32_t pw = posbits[w];
        uint32_t aw = augbits[w];
        uint32_t xw = xbits[w];
        p = (uint32_t)__popc(pw);
        f = (uint32_t)__popc(xw & ~aw);
        n = (uint32_t)__popc(~aw);
    }
    #pragma unroll
    for (int off = 4; off > 0; off >>= 1) {
        p += __shfl_down(p, off);
        f += __shfl_down(f, off);
        n += __shfl_down(n, off);
    }
    if (lane == 0) { posCnt[b] = p; fpCnt[b] = f; negCnt[b] = n; }
}

// ---------------------------------------------------------------------------
// K4: per-row scan of the 1024 block counts. Decides fp-vs-neg selection,
// produces block offsets and row totals. One 256-thread block per row.
// Packed u64 Hillis-Steele scan (pos in low 32, chosen in high 32).
// ---------------------------------------------------------------------------
__global__ void kscan(const uint32_t* __restrict__ posCnt,
                      const uint32_t* __restrict__ fpCnt,
                      const uint32_t* __restrict__ negCnt,
                      uint32_t* __restrict__ posOff,
                      uint32_t* __restrict__ chOff,
                      uint32_t* __restrict__ rowMeta) {
    int r   = blockIdx.x;
    int tid = threadIdx.x;
    __shared__ uint64_t s[256];
    __shared__ uint32_t flagSh;

    // Phase 1: total false-positive count -> selection flag
    uint32_t fN = 0;
    #pragma unroll
    for (int j = 0; j < 4; ++j) fN += fpCnt[r * NBLK + tid * 4 + j];
    s[tid] = fN;
    __syncthreads();
    for (int off = 128; off > 0; off >>= 1) {
        if (tid < off) s[tid] += s[tid + off];
        __syncthreads();
    }
    if (tid == 0) flagSh = ((uint32_t)s[0] > 0u) ? 1u : 0u;
    __syncthreads();
    bool useFp = (flagSh != 0u);

    // Phase 2: exclusive scan of (pos, chosen) counts
    uint32_t p[4], c[4], tp = 0, tc = 0;
    #pragma unroll
    for (int j = 0; j < 4; ++j) {
        int b = r * NBLK + tid * 4 + j;
        p[j] = posCnt[b];
        c[j] = useFp ? fpCnt[b] : negCnt[b];
        tp += p[j];
        tc += c[j];
    }
    uint64_t v = (uint64_t)tp | ((uint64_t)tc << 32);
    s[tid] = v;
    __syncthreads();
    for (int off = 1; off < 256; off <<= 1) {
        uint64_t add = (tid >= off) ? s[tid - off] : 0ull;
        __syncthreads();
        s[tid] += add;
        __syncthreads();
    }
    uint64_t incl = s[tid];
    uint64_t excl = incl - v;
    uint32_t rp = (uint32_t)excl;
    uint32_t rc = (uint32_t)(excl >> 32);
    #pragma unroll
    for (int j = 0; j < 4; ++j) {
        posOff[r * NBLK + tid * 4 + j] = rp; rp += p[j];
        chOff [r * NBLK + tid * 4 + j] = rc; rc += c[j];
    }
    if (tid == 255) {
        rowMeta[r * 4 + 0] = (uint32_t)incl;          // cntP
        rowMeta[r * 4 + 1] = (uint32_t)(incl >> 32);  // cntC
        rowMeta[r * 4 + 2] = useFp ? 1u : 0u;
    }
}

// ---------------------------------------------------------------------------
// K5: stream compaction. One wave == one mask word, so the mask word itself
// is the ballot; local rank = popcount of lower lane bits.
// ---------------------------------------------------------------------------
__global__ void kcompact(const float* __restrict__ x,
                         const uint32_t* __restrict__ posbits,
                         const uint32_t* __restrict__ xbits,
                         const uint32_t* __restrict__ augbits,
                         const uint32_t* __restrict__ posOff,
                         const uint32_t* __restrict__ chOff,
                         const uint32_t* __restrict__ rowMeta,
                         float* __restrict__ XP, float* __restrict__ XC) {
    int g    = blockIdx.x * 256 + threadIdx.x;
    int r    = g >> 18;                 // / L_IMG
    int blk  = (g >> 8) & (NBLK - 1);
    int word = g >> 5;
    int lane = threadIdx.x & 31;
    int wave = threadIdx.x >> 5;

    uint32_t pw = posbits[word];
    uint32_t aw = augbits[word];
    uint32_t xw = xbits[word];
    bool useFp  = rowMeta[r * 4 + 2] != 0u;
    uint32_t cw = useFp ? (xw & ~aw) : ~aw;

    __shared__ uint32_t wTotP[8], wTotC[8];
    if (lane == 0) { wTotP[wave] = (uint32_t)__popc(pw); wTotC[wave] = (uint32_t)__popc(cw); }
    __syncthreads();
    uint32_t offP = 0, offC = 0;
    for (int wv = 0; wv < wave; ++wv) { offP += wTotP[wv]; offC += wTotC[wv]; }

    uint32_t lmask = (lane == 0) ? 0u : (0xFFFFFFFFu >> (32 - lane));
    float xv = x[g];
    size_t rowBase = (size_t)r * L_IMG;
    if ((pw >> lane) & 1u)
        XP[rowBase + posOff[r * NBLK + blk] + offP + (uint32_t)__popc(pw & lmask)] = xv;
    if ((cw >> lane) & 1u)
        XC[rowBase + chOff[r * NBLK + blk] + offC + (uint32_t)__popc(cw & lmask)] = xv;
}

// ---------------------------------------------------------------------------
// K6: main eval. Stages x and t chunks into LDS via CDNA5 async-to-LDS loads
// (ASYNCcnt path), gathers tiled dup_pos/dup_fp from the compacted arrays,
// accumulates the three softplus sums, fixed-tree block reduction.
// ---------------------------------------------------------------------------
__global__ void keval(const float* __restrict__ x, const float* __restrict__ t,
                      const uint32_t* __restrict__ rowMeta,
                      const float* __restrict__ XP, const float* __restrict__ XC,
                      float* __restrict__ partials) {
    int r     = blockIdx.x >> 8;          // 256 chunks per row
    int chunk = blockIdx.x & 255;
    int tid   = threadIdx.x;
    int lane  = tid & 31;
    int wave  = tid >> 5;
    size_t base = (size_t)r * L_IMG + (size_t)chunk * CH;

    __shared__ __align__(16) float xs[CH];
    __shared__ __align__(16) float tsh[CH];

    // CDNA5 async global->LDS staging (GLOBAL_LOAD_ASYNC_TO_LDS_B128, ASYNCcnt)
    unsigned ldx = (unsigned)(uintptr_t)(&xs[tid * 4]);
    unsigned ldt = (unsigned)(uintptr_t)(&tsh[tid * 4]);
    const float* gx = x + base + (size_t)tid * 4;
    const float* gt = t + base + (size_t)tid * 4;
    asm volatile("global_load_async_to_lds_b128 %0, %1, off" :: "v"(ldx), "v"(gx) : "memory");
    asm volatile("global_load_async_to_lds_b128 %0, %1, off" :: "v"(ldt), "v"(gt) : "memory");
    asm volatile("s_wait_asynccnt 0x0" ::: "memory");
    __syncthreads();

    uint32_t cntP = rowMeta[r * 4 + 0];
    uint32_t cntC = rowMeta[r * 4 + 1];
    const float* XPr = XP + (size_t)r * L_IMG;
    const float* XCr = XC + (size_t)r * L_IMG;

    uint32_t iL = (uint32_t)(chunk * CH + tid * 4);   // index within row
    uint32_t kp = cntP ? (iL % cntP) : 0u;
    uint32_t kc = cntC ? (iL % cntC) : 0u;

    float s1 = 0.0f, s2 = 0.0f, s3 = 0.0f;
    #pragma unroll
    for (int j = 0; j < 4; ++j) {
        float xv = xs[tid * 4 + j];
        float tv = tsh[tid * 4 + j];
        float dp  = cntP ? XPr[kp] : 5.0f;
        float dfp = cntC ? XCr[kc] : -5.0f;
        if (cntP) { if (++kp == cntP) kp = 0u; }
        if (cntC) { if (++kc == cntC) kc = 0u; }
        float sim = dp * xv;
        s1 += softplusf(sim) - sim * tv;
        s2 += softplusf(dp) - dp;
        s3 += softplusf(dp * dfp);
    }

    // wave32 tree reduce, then cross-wave via LDS (fixed order -> deterministic)
    #pragma unroll
    for (int off = 16; off > 0; off >>= 1) {
        s1 += __shfl_down(s1, off);
        s2 += __shfl_down(s2, off);
        s3 += __shfl_down(s3, off);
    }
    __shared__ float r1[8], r2[8], r3[8];
    if (lane == 0) { r1[wave] = s1; r2[wave] = s2; r3[wave] = s3; }
    __syncthreads();
    if (tid == 0) {
        float a = 0.0f, b = 0.0f, c = 0.0f;
        #pragma unroll
        for (int wv = 0; wv < 8; ++wv) { a += r1[wv]; b += r2[wv]; c += r3[wv]; }
        partials[blockIdx.x * 3 + 0] = a;
        partials[blockIdx.x * 3 + 1] = b;
        partials[blockIdx.x * 3 + 2] = c;
    }
}

// ---------------------------------------------------------------------------
// K7: final deterministic double-precision reduction of per-block partials.
// ---------------------------------------------------------------------------
__global__ void kfinal(const float* __restrict__ partials, float* __restrict__ out) {
    int tid = threadIdx.x;
    __shared__ double d1[256], d2[256], d3[256];
    double a = 0.0, b = 0.0, c = 0.0;
    for (int i = tid; i < NEVAL; i += 256) {
        a += (double)partials[i * 3 + 0];
        b += (double)partials[i * 3 + 1];
        c += (double)partials[i * 3 + 2];
    }
    d1[tid] = a; d2[tid] = b; d3[tid] = c;
    __syncthreads();
    for (int off = 128; off > 0; off >>= 1) {
        if (tid < off) { d1[tid] += d1[tid + off]; d2[tid] += d2[tid + off]; d3[tid] += d3[tid + off]; }
        __syncthreads();
    }
    if (tid == 0) {
        const double N = (double)R_IMG * (double)L_IMG;   // 12582912
        double res = (d1[0] + d2[0] + 0.1 * d3[0]) / N;
        out[0] = (float)res;
    }
}

// ---------------------------------------------------------------------------
// Host launcher
// ---------------------------------------------------------------------------
extern "C" void kernel_launch(void* const* d_in, const int* in_sizes, int n_in,
                              void* d_out, int out_size, void* d_ws, size_t ws_size,
                              hipStream_t stream) {
    (void)in_sizes; (void)n_in; (void)out_size; (void)ws_size;
    const float* x = (const float*)d_in[0];   // "input"
    const float* t = (const float*)d_in[1];   // "target"
    float* out = (float*)d_out;

    char* ws = (char*)d_ws;
    size_t o = 0;
    auto take = [&](size_t bytes) -> char* {
        char* p = ws + o;
        o = (o + bytes + 255) & ~(size_t)255;
        return p;
    };
    uint32_t* rowMeta  = (uint32_t*)take((size_t)R_IMG * 4 * sizeof(uint32_t));
    uint32_t* posCnt   = (uint32_t*)take((size_t)TOTBLK * sizeof(uint32_t));
    uint32_t* fpCnt    = (uint32_t*)take((size_t)TOTBLK * sizeof(uint32_t));
    uint32_t* negCnt   = (uint32_t*)take((size_t)TOTBLK * sizeof(uint32_t));
    uint32_t* posOff   = (uint32_t*)take((size_t)TOTBLK * sizeof(uint32_t));
    uint32_t* chOff    = (uint32_t*)take((size_t)TOTBLK * sizeof(uint32_t));
    uint32_t* posbits  = (uint32_t*)take((size_t)R_IMG * WPI * sizeof(uint32_t));
    uint32_t* xbits    = (uint32_t*)take((size_t)R_IMG * WPI * sizeof(uint32_t));
    uint32_t* augbits  = (uint32_t*)take((size_t)R_IMG * WPI * sizeof(uint32_t));
    float*    XP       = (float*)   take((size_t)R_IMG * L_IMG * sizeof(float));
    float*    XC       = (float*)   take((size_t)R_IMG * L_IMG * sizeof(float));
    float*    partials = (float*)   take((size_t)NEVAL * 3 * sizeof(float));

    kpack   <<<TOTBLK,            256, 0, stream>>>(x, t, posbits, xbits);
    kdilate <<<(R_IMG * WPI)/256, 256, 0, stream>>>(posbits, augbits);
    kcount  <<<TOTBLK,             32, 0, stream>>>(posbits, xbits, augbits, posCnt, fpCnt, negCnt);
    kscan   <<<R_IMG,             256, 0, stream>>>(posCnt, fpCnt, negCnt, posOff, chOff, rowMeta);
    kcompact<<<TOTBLK,            256, 0, stream>>>(x, posbits, xbits, augbits, posOff, chOff, rowMeta, XP, XC);
    keval   <<<NEVAL,             256, 0, stream>>>(x, t, rowMeta, XP, XC, partials);
    kfinal  <<<1,                 256, 0, stream>>>(partials, out);
}